// all2vec_41437844472386
// MI455X (gfx1250) — compile-verified
//
#include <hip/hip_runtime.h>

// node2vec-style negative-sampling loss, MI455X (gfx1250, wave32).
//
// Strategy: gather-bound problem (~109 MB @ 23.3 TB/s => ~4.7us floor).
// One wave handles 16 batch elements. Scores computed with full-precision
// V_WMMA_F32_16X16X4_F32 tiles: A = 16 target rows (emb[pos_v]),
// B = one of 12 query rows per batch element (u1, u2, neg0..9); the tile
// diagonal is the 16 wanted dot products. A stays in VGPRs across all 12
// tiles so every embedding row is read from HBM exactly once.

typedef __attribute__((ext_vector_type(2))) float v2f;
typedef __attribute__((ext_vector_type(8))) float v8f;

#define NDIM 128
#define NNEG 10
#define NQ   12   // u1, u2, 10 negatives

__device__ __forceinline__ float log_sigmoid(float x) {
    // log_sigmoid(x) = -log1p(exp(-x)), numerically stable form
    return fminf(x, 0.0f) - log1pf(expf(-fabsf(x)));
}

__global__ __launch_bounds__(256)
void n2v_wmma_kernel(const int* __restrict__ pos_v,
                     const int* __restrict__ pos_u,
                     const int* __restrict__ neg,
                     const float* __restrict__ weights,
                     const float* __restrict__ emb,
                     const float* __restrict__ emb_ctx,
                     float* __restrict__ out,
                     int batch)
{
    const int lane = threadIdx.x & 31;
    const int wave = (int)((blockIdx.x * blockDim.x + threadIdx.x) >> 5);
    const int b0   = wave * 16;
    if (b0 >= batch) return;               // wave-uniform: EXEC stays all-ones

    const int m    = lane & 15;            // A-row / B-col owned by this lane
    const int kofs = (lane >> 4) * 2;      // f32 WMMA layout: hi lanes hold K+2,K+3
    int bm = b0 + m;
    if (bm >= batch) bm = batch - 1;       // safe clamp (batch % 16 == 0 normally)

    // Per-lane gather base pointers (row = 128 f32 = 512 B).
    const float* rowA = emb + (size_t)pos_v[bm] * NDIM;
    const float* rowB[NQ];
    rowB[0] = emb     + (size_t)pos_u[bm] * NDIM;
    rowB[1] = emb_ctx + (size_t)pos_u[bm] * NDIM;
#pragma unroll
    for (int k = 0; k < NNEG; ++k)
        rowB[2 + k] = emb_ctx + (size_t)neg[bm * NNEG + k] * NDIM;

    v8f acc[NQ];
#pragma unroll
    for (int t = 0; t < NQ; ++t) acc[t] = (v8f){};

    // Sweep K = 128 in 32 chunks of 4; A chunk reused by all 12 WMMA tiles.
#pragma unroll 4
    for (int c = 0; c < NDIM / 4; ++c) {
        const int kk = 4 * c + kofs;
        v2f a = *(const v2f*)(rowA + kk);
#pragma unroll
        for (int t = 0; t < NQ; ++t) {
            v2f bb = *(const v2f*)(rowB[t] + kk);
            // 8 args: (neg_a, A, neg_b, B, c_mod, C, reuse_a, reuse_b)
            acc[t] = __builtin_amdgcn_wmma_f32_16x16x4_f32(
                false, a, false, bb, (short)0, acc[t], false, false);
        }
    }

    // C/D f32 16x16 layout: lane<16 -> (M=vgpr, N=lane); lane>=16 -> (M=vgpr+8, N=lane-16).
    // Diagonal m: lanes 0..7 at vgpr=lane; lanes 24..31 at vgpr=lane-24.
    const bool active = (lane < 8) || (lane >= 24);
    if (!active) return;
    const int r   = lane & 7;
    const int myb = b0 + ((lane < 8) ? lane : (lane - 16));
    if (myb >= batch) return;

    float s[NQ];
#pragma unroll
    for (int t = 0; t < NQ; ++t) {
        float x = acc[t][0];
        x = (r == 1) ? acc[t][1] : x;
        x = (r == 2) ? acc[t][2] : x;
        x = (r == 3) ? acc[t][3] : x;
        x = (r == 4) ? acc[t][4] : x;
        x = (r == 5) ? acc[t][5] : x;
        x = (r == 6) ? acc[t][6] : x;
        x = (r == 7) ? acc[t][7] : x;
        s[t] = x;
    }

    const float w = weights[myb];
    float neg_sum = 0.0f;
#pragma unroll
    for (int k = 0; k < NNEG; ++k)
        neg_sum += log_sigmoid(-s[2 + k]);

    out[myb]         = w * (-log_sigmoid(s[0]) - neg_sum);  // score_1
    out[batch + myb] = w * (-log_sigmoid(s[1]) - neg_sum);  // score_2
}

extern "C" void kernel_launch(void* const* d_in, const int* in_sizes, int n_in,
                              void* d_out, int out_size, void* d_ws, size_t ws_size,
                              hipStream_t stream) {
    const int*   pos_v   = (const int*)d_in[0];
    const int*   pos_u   = (const int*)d_in[1];
    const int*   neg     = (const int*)d_in[2];
    const float* weights = (const float*)d_in[3];
    const float* emb     = (const float*)d_in[4];
    const float* emb_ctx = (const float*)d_in[5];
    float*       out     = (float*)d_out;

    const int batch   = in_sizes[0];            // 16384
    const int waves   = (batch + 15) / 16;      // 16 batch elements per wave32
    const int threads = waves * 32;
    const int block   = 256;                    // 8 waves / block
    const int grid    = (threads + block - 1) / block;

    n2v_wmma_kernel<<<grid, block, 0, stream>>>(pos_v, pos_u, neg, weights,
                                                emb, emb_ctx, out, batch);
}